// GraphConv_34668976013388
// MI455X (gfx1250) — compile-verified
//
#include <hip/hip_runtime.h>
#include <hip/hip_bf16.h>

// GraphConv on MI455X (gfx1250):
//   h = x @ W            (fp32 WMMA 16x16x4, memory-bound -> keep full fp32)
//   out = bias           (init)
//   out[dst] += ew*h[src] (wave-per-edge, float4 lanes, L2 atomics)

typedef __attribute__((ext_vector_type(2))) float v2f;
typedef __attribute__((ext_vector_type(8))) float v8f;

#define IN_F   128
#define OUT_F  128
#define N_NODES 50000
#define N_EDGES 800000

// ---------------------------------------------------------------------------
// Kernel 1: h[N,128] = x[N,128] @ W[128,128] using V_WMMA_F32_16X16X4_F32.
// Block = 256 threads (8 wave32). Block owns rows [16*b, 16*b+16).
// Wave w owns the 16x16 output tile at columns [16*w, 16*w+16).
// X strip (16x128 = 8KB) staged in LDS, shared by all 8 waves.
// ---------------------------------------------------------------------------
__global__ __launch_bounds__(256) void gemm_h_kernel(
    const float* __restrict__ x, const float* __restrict__ w,
    float* __restrict__ h)
{
    __shared__ float xs[16 * IN_F];

    const int tid  = threadIdx.x;
    const int row0 = blockIdx.x * 16;

    // Cooperative load of the 16x128 X strip: 2048 floats = 512 float4.
    {
        const float4* xg  = (const float4*)(x + (size_t)row0 * IN_F);
        float4*       xs4 = (float4*)xs;
        xs4[tid]       = xg[tid];
        xs4[tid + 256] = xg[tid + 256];
    }
    __syncthreads();

    const int wave = tid >> 5;
    const int lane = tid & 31;
    const int half = (lane >> 4) & 1;   // lane group: 0 -> K/M low, 1 -> high
    const int mn   = lane & 15;         // M for A-layout, N for B-layout
    const int col0 = wave * 16;

    v8f acc = {};

    // K = 128 -> 32 chained WMMAs, fully unrolled.
    // A layout (16x4 f32): lanes 0-15 hold K=0,1 in v0,v1; lanes 16-31 K=2,3.
    // B layout (4x16 f32): lanes 0-15 hold N=0..15 at K=0,1; lanes 16-31 K=2,3.
    #pragma unroll
    for (int k0 = 0; k0 < IN_F; k0 += 4) {
        const int k = k0 + 2 * half;
        v2f a = *(const v2f*)(xs + mn * IN_F + k);        // x[row0+mn][k..k+1]
        v2f b;
        b.x = w[(k + 0) * OUT_F + col0 + mn];             // W[k  ][col0+mn]
        b.y = w[(k + 1) * OUT_F + col0 + mn];             // W[k+1][col0+mn]
        acc = __builtin_amdgcn_wmma_f32_16x16x4_f32(
            /*neg_a=*/false, a, /*neg_b=*/false, b,
            /*c_mod=*/(short)0, acc, /*reuse_a=*/false, /*reuse_b=*/false);
    }

    // C/D layout: VGPR r, lanes 0-15 -> M=r, lanes 16-31 -> M=r+8; N = lane&15.
    float* hp = h + (size_t)(row0 + half * 8) * OUT_F + col0 + mn;
    #pragma unroll
    for (int r = 0; r < 8; ++r)
        hp[(size_t)r * OUT_F] = acc[r];
}

// ---------------------------------------------------------------------------
// Kernel 2: out[n][f] = bias[f]   (float4 vectorized; 128 floats = 32 float4)
// ---------------------------------------------------------------------------
__global__ __launch_bounds__(256) void init_out_kernel(
    const float* __restrict__ bias, float* __restrict__ out, int total4)
{
    const int i = blockIdx.x * blockDim.x + threadIdx.x;
    if (i < total4) {
        const float4* b4 = (const float4*)bias;
        ((float4*)out)[i] = b4[i & 31];
    }
}

// ---------------------------------------------------------------------------
// Kernel 3: one wave32 per edge (grid-stride). Each lane handles 4 contiguous
// floats of the 128-wide feature row: coalesced 512B gather of h[src] (L2-hot,
// h = 25.6MB << 192MB L2), scale by edge weight, 4x global_atomic_add_f32
// into out[dst] (resolved at the L2 atomic units).
// ---------------------------------------------------------------------------
__global__ __launch_bounds__(256) void edge_scatter_kernel(
    const float* __restrict__ h, const float* __restrict__ ew,
    const int* __restrict__ src, const int* __restrict__ dst,
    float* __restrict__ out, int n_edges)
{
    const int lane   = threadIdx.x & 31;
    const int gwave  = (blockIdx.x * blockDim.x + threadIdx.x) >> 5;
    const int nwaves = (gridDim.x * blockDim.x) >> 5;

    for (int e = gwave; e < n_edges; e += nwaves) {
        const int   s = src[e];
        const int   d = dst[e];
        const float wgt = ew[e];

        const float4 hv = *(const float4*)(h + (size_t)s * OUT_F + lane * 4);
        float* o = out + (size_t)d * OUT_F + lane * 4;
        atomicAdd(o + 0, hv.x * wgt);
        atomicAdd(o + 1, hv.y * wgt);
        atomicAdd(o + 2, hv.z * wgt);
        atomicAdd(o + 3, hv.w * wgt);
    }
}

// ---------------------------------------------------------------------------
// Launch
// ---------------------------------------------------------------------------
extern "C" void kernel_launch(void* const* d_in, const int* in_sizes, int n_in,
                              void* d_out, int out_size, void* d_ws, size_t ws_size,
                              hipStream_t stream)
{
    const float* x    = (const float*)d_in[0];   // [N, 128]
    const float* wgt  = (const float*)d_in[1];   // [128, 128]
    const float* bias = (const float*)d_in[2];   // [1, 128]
    const float* ew   = (const float*)d_in[3];   // [E]
    const int*   esrc = (const int*)d_in[4];     // [E]
    const int*   edst = (const int*)d_in[5];     // [E]
    float*       out  = (float*)d_out;           // [N, 128]

    const int n_nodes = in_sizes[0] / IN_F;      // 50000
    const int n_edges = in_sizes[3];             // 800000

    float* h = (float*)d_ws;                     // [N, 128] scratch (25.6 MB)

    // 1) h = x @ W  (WMMA fp32). 50000/16 = 3125 row strips.
    gemm_h_kernel<<<n_nodes / 16, 256, 0, stream>>>(x, wgt, h);

    // 2) out = bias
    const int total4 = n_nodes * OUT_F / 4;
    init_out_kernel<<<(total4 + 255) / 256, 256, 0, stream>>>(bias, out, total4);

    // 3) out[dst] += ew * h[src]
    edge_scatter_kernel<<<8192, 256, 0, stream>>>(h, ew, esrc, edst, out, n_edges);
}